// PatchShift_27968827032243
// MI455X (gfx1250) — compile-verified
//
#include <hip/hip_runtime.h>

// ---------------------------------------------------------------------------
// Temporal Patch Shift (forward): out[b,t,h,w,:] = x[b,(t - S[h%3][w%3])%8,h,w,:]
// Zero-FLOP permute => HBM-bandwidth bound: 154 MB read + 154 MB write
//   => ~13 us ceiling @ 23.3 TB/s on MI455X. No WMMA path applies.
//
// Key structure: for fixed (b,h,w) the shift s is constant over t, and the
// T-permutation is a rotation — the 8 source rows ARE the 8 destination rows.
// One block owns all T=8 rows of one (b,h,w):
//   - blockIdx -> (b,h,w): ONE scalar div/mod chain per block (not per row).
//   - Loads iterate source t' = j (compile-time): byte offset j*602112 fits
//     the signed 24-bit instruction offset => literal-offset b128 loads off a
//     single address register.
//   - Stores go to t = (j + s) & 7: one scalar add per row.
// Each lane moves one float4 per row: 8 NT b128 loads + 8 NT b128 stores.
// ---------------------------------------------------------------------------

typedef __attribute__((ext_vector_type(4))) float v4f;

namespace {
constexpr int  kB   = 32;
constexpr int  kT   = 8;
constexpr int  kH   = 14;
constexpr int  kW   = 14;
constexpr int  kC   = 768;
constexpr int  kC4  = kC / 4;                 // 192 float4 per row
constexpr int  kHW  = kH * kW;                // 196 rows per (b,t) slice
constexpr long kTStride4 = (long)kHW * kC4;   // 37,632 float4 between t-slices
constexpr int  kBlocks = kB * kH * kW;        // 6272 (b,h,w) triples
// Shift table S[h%3][w%3] = {{-4,1,2},{-1,0,3},{-2,-3,4}}, packed as (s+4)
// nibbles; entry k = (h%3)*3 + (w%3) lives in bits [4k+3 : 4k].
constexpr unsigned long long kShiftNibbles = 0x812743650ULL;
}

__global__ __launch_bounds__(kC4) void PatchShift_tps_kernel(
    const v4f* __restrict__ x, v4f* __restrict__ out) {
  const int lane = threadIdx.x;               // 0..191: one float4 of each row

  // ---- once per block: decompose (b,h,w) and look up shift (all SALU) ----
  const int r  = blockIdx.x;                  // 0..6271
  const int w  = r % kW;
  const int hb = r / kW;
  const int h  = hb % kH;
  const int b  = hb / kH;

  const int kk = (h % 3) * 3 + (w % 3);
  const int s  = (int)((kShiftNibbles >> (4 * kk)) & 0xF) - 4;   // s in [-4,4]

  // Row id at t=0 for this (b,h,w); per-lane float4 base index.
  const long rowBase = (long)b * (kT * kHW) + h * kW + w;
  const long base4   = rowBase * kC4 + lane;

  const v4f* __restrict__ src = x + base4;
  v4f* __restrict__       dst = out + base4;

  // ---- 8 rows: loads use compile-time immediate offsets, stores rotated ----
#pragma unroll
  for (int j = 0; j < kT; ++j) {              // j = source t'
    const v4f v = __builtin_nontemporal_load(src + j * kTStride4);
    const int tdst = (j + s + kT) & (kT - 1); // destination t = (t' + s) mod 8
    __builtin_nontemporal_store(v, dst + (long)tdst * kTStride4);
  }
}

extern "C" void kernel_launch(void* const* d_in, const int* in_sizes, int n_in,
                              void* d_out, int out_size, void* d_ws, size_t ws_size,
                              hipStream_t stream) {
  (void)in_sizes; (void)n_in; (void)out_size; (void)d_ws; (void)ws_size;
  const v4f* x   = (const v4f*)d_in[0];
  v4f*       out = (v4f*)d_out;

  PatchShift_tps_kernel<<<kBlocks, kC4, 0, stream>>>(x, out);
}